// StreamingAcousticDecoderWrapper_49864570307000
// MI455X (gfx1250) — compile-verified
//
#include <hip/hip_runtime.h>
#include <math.h>

typedef float v2f __attribute__((ext_vector_type(2)));
typedef float v8f __attribute__((ext_vector_type(8)));

// ---------------------------------------------------------------------------
// Small helpers
// ---------------------------------------------------------------------------
__device__ __forceinline__ float gelu_exact(float v) {
  // exact GELU: 0.5*x*(1+erf(x/sqrt(2)))
  return 0.5f * v * (1.0f + erff(v * 0.70710678118654752440f));
}

// ---------------------------------------------------------------------------
// cache slice writer: dst[c*ctx + j] = src[c*T + (T-ctx+j)]
// ---------------------------------------------------------------------------
__global__ void k_copy_cache(const float* __restrict__ src, float* __restrict__ dst,
                             int C, int T, int ctx) {
  int i = blockIdx.x * blockDim.x + threadIdx.x;
  if (i >= C * ctx) return;
  int c = i / ctx, j = i % ctx;
  dst[i] = src[c * T + (T - ctx + j)];
}

// ---------------------------------------------------------------------------
// Stage-0 standard conv: w (cout,cin,k), full_in = [cache(ctx) | x(T)], Tout=T
// ---------------------------------------------------------------------------
__global__ void k_conv_up(const float* __restrict__ x, const float* __restrict__ cache,
                          const float* __restrict__ w, const float* __restrict__ b,
                          float* __restrict__ y, int cin, int cout, int K, int ctx, int T) {
  int i = blockIdx.x * blockDim.x + threadIdx.x;
  if (i >= cout * T) return;
  int co = i / T, t = i % T;
  float acc = b[co];
  for (int ci = 0; ci < cin; ++ci) {
    const float* wr = w + (co * cin + ci) * K;
    for (int kk = 0; kk < K; ++kk) {
      int ti = t + kk;
      float v = (ti < ctx) ? cache[ci * ctx + ti] : x[ci * T + (ti - ctx)];
      acc += v * wr[kk];
    }
  }
  y[i] = acc;
}

// ---------------------------------------------------------------------------
// Transposed conv (ctx==1, K % S == 0). After the reference's crop-and-tail-
// slice, output col t' maps to full_out col t'+S and has exactly K/S taps:
//   tap j: t_in = t'/S + 1 - j (into full_in), kk = (t'%S) + j*S
// w layout (cin,cout,K): w[(ci*cout+co)*K + kk]
// ---------------------------------------------------------------------------
__global__ void k_convtr(const float* __restrict__ x, const float* __restrict__ cache,
                         const float* __restrict__ w, const float* __restrict__ b,
                         float* __restrict__ y, int cin, int cout, int K, int S,
                         int Tin, int Tout) {
  int i = blockIdx.x * blockDim.x + threadIdx.x;
  if (i >= cout * Tout) return;
  int co = i / Tout, t = i % Tout;
  int p = t % S, base = t / S;
  int ntap = K / S;
  float acc = b[co];
  for (int ci = 0; ci < cin; ++ci) {
    const float* wr = w + (ci * cout + co) * K + p;
    for (int j = 0; j < ntap; ++j) {
      int ti = base + 1 - j;  // index into full_in (ctx=1)
      float v = (ti == 0) ? cache[ci] : x[ci * Tin + (ti - 1)];
      acc += v * wr[j * S];
    }
  }
  y[i] = acc;
}

// ---------------------------------------------------------------------------
// Channel LayerNorm at each timestep t (two-pass for numerical fidelity)
// grid = T blocks of 128 threads
// ---------------------------------------------------------------------------
__global__ void k_ln(const float* __restrict__ x, const float* __restrict__ w,
                     const float* __restrict__ b, float* __restrict__ y, int C, int T) {
  int t = blockIdx.x;
  __shared__ float sh[128];
  float s = 0.f;
  for (int c = threadIdx.x; c < C; c += 128) s += x[c * T + t];
  sh[threadIdx.x] = s; __syncthreads();
  for (int o = 64; o > 0; o >>= 1) {
    if (threadIdx.x < o) sh[threadIdx.x] += sh[threadIdx.x + o];
    __syncthreads();
  }
  float m = sh[0] / (float)C;
  __syncthreads();
  float q = 0.f;
  for (int c = threadIdx.x; c < C; c += 128) { float d = x[c * T + t] - m; q += d * d; }
  sh[threadIdx.x] = q; __syncthreads();
  for (int o = 64; o > 0; o >>= 1) {
    if (threadIdx.x < o) sh[threadIdx.x] += sh[threadIdx.x + o];
    __syncthreads();
  }
  float r = rsqrtf(sh[0] / (float)C + 1e-6f);
  for (int c = threadIdx.x; c < C; c += 128)
    y[c * T + t] = (x[c * T + t] - m) * r * w[c] + b[c];
}

// ---------------------------------------------------------------------------
// Depthwise conv block (k=7, ctx=6), fused gamma + residual, in-place on x.
// ---------------------------------------------------------------------------
__global__ void k_dwconv(float* __restrict__ x, const float* __restrict__ xn,
                         const float* __restrict__ cache, const float* __restrict__ w,
                         const float* __restrict__ b, const float* __restrict__ g,
                         int C, int T) {
  int i = blockIdx.x * blockDim.x + threadIdx.x;
  if (i >= C * T) return;
  int c = i / T, t = i % T;
  float acc = b[c];
  const float* wr = w + c * 7;
#pragma unroll
  for (int kk = 0; kk < 7; ++kk) {
    int ti = t + kk;
    float v = (ti < 6) ? cache[c * 6 + ti] : xn[c * T + (ti - 6)];
    acc += v * wr[kk];
  }
  x[i] = x[i] + g[c] * acc;
}

// ---------------------------------------------------------------------------
// WMMA fp32 GEMM:  D[m,n] = epi( sum_k Wcm[k*M + m] * B[k*T + n] )
//   MODE 0: D = gelu(acc + bias[m])
//   MODE 1: D[m,n] = D[m,n] + gamma[m]*(acc + bias[m])   (in-place residual)
//
// FAST path (T % 64 == 0, all stages except stage 0):
//   block = 128 thr (4 waves); wave computes a 32x64 macro-tile = 2(M) x 4(N)
//   16x16 WMMA tiles. Per K-step (K+=4): 4 A loads + 8 B loads -> 8 v_wmma
//   (1.5 loads/wmma, 8 independent accumulator chains). No bounds checks, no
//   divergence -> EXEC all-1s for every WMMA. Requires M % 128 == 0 (true).
//
// Fragment layouts (ISA 7.12.2):
//   A: lane l elem v -> A[M = l%16, K = (l/16)*2 + v]
//   B: lane l elem v -> B[K = (l/16)*2 + v, N = l%16]
//   D: elem r        -> D[M = r + 8*(l/16), N = l%16]
// ---------------------------------------------------------------------------
template <int MODE>
__global__ void k_gemm_wmma_fast(const float* __restrict__ Wcm, const float* __restrict__ B,
                                 float* D, const float* __restrict__ bias,
                                 const float* __restrict__ gamma, int M, int K, int T) {
  int lane = threadIdx.x & 31;
  int wave = threadIdx.x >> 5;
  int half = lane >> 4;
  int mn = lane & 15;
  int M0 = blockIdx.x * 128 + wave * 32;  // two M tiles: M0, M0+16
  int N0 = blockIdx.y * 64;               // four N tiles: N0 + 16*j

  const v8f vzero = {0.f, 0.f, 0.f, 0.f, 0.f, 0.f, 0.f, 0.f};
  v8f acc[2][4];
#pragma unroll
  for (int i = 0; i < 2; ++i)
#pragma unroll
    for (int j = 0; j < 4; ++j) acc[i][j] = vzero;

  const float* ap = Wcm + (half * 2) * M + M0 + mn;  // A[M0+mn, half*2 + v]
  const float* bp = B + (half * 2) * T + N0 + mn;    // B[half*2 + v, N0+mn]
  const int aStep = 4 * M, bStep = 4 * T;

  for (int k0 = 0; k0 < K; k0 += 4) {
    v2f a0, a1;
    a0[0] = ap[0];
    a0[1] = ap[M];
    a1[0] = ap[16];
    a1[1] = ap[M + 16];
    v2f bb[4];
#pragma unroll
    for (int j = 0; j < 4; ++j) {
      bb[j][0] = bp[16 * j];
      bb[j][1] = bp[T + 16 * j];
    }
#pragma unroll
    for (int j = 0; j < 4; ++j) {
      acc[0][j] = __builtin_amdgcn_wmma_f32_16x16x4_f32(false, a0, false, bb[j],
                                                        (short)0, acc[0][j], false, false);
      acc[1][j] = __builtin_amdgcn_wmma_f32_16x16x4_f32(false, a1, false, bb[j],
                                                        (short)0, acc[1][j], false, false);
    }
    ap += aStep;
    bp += bStep;
  }

#pragma unroll
  for (int i = 0; i < 2; ++i) {
    int mt = M0 + i * 16;
#pragma unroll
    for (int j = 0; j < 4; ++j) {
      int n = N0 + 16 * j + mn;
#pragma unroll
      for (int r = 0; r < 8; ++r) {
        int m = mt + r + half * 8;
        int idx = m * T + n;
        float v = acc[i][j][r] + bias[m];
        if (MODE == 0) D[idx] = gelu_exact(v);
        else           D[idx] = D[idx] + gamma[m] * v;
      }
    }
  }
}

// Guarded reference path: one 16x16 tile per wave, N bounds-checked.
// Used only for stage 0 (T = 8).
template <int MODE>
__global__ void k_gemm_wmma_ref(const float* __restrict__ Wcm, const float* __restrict__ B,
                                float* D, const float* __restrict__ bias,
                                const float* __restrict__ gamma, int M, int K, int T) {
  int lane = threadIdx.x & 31;
  int wave = threadIdx.x >> 5;
  int M0 = (blockIdx.x * 4 + wave) * 16;
  int N0 = blockIdx.y * 16;
  int half = lane >> 4;
  int mn = lane & 15;
  int n = N0 + mn;
  bool nin = (n < T);

  v8f acc = {0.f, 0.f, 0.f, 0.f, 0.f, 0.f, 0.f, 0.f};
  const float* Wp = Wcm + M0 + mn;
  const float* Bp = B + (nin ? n : 0);

  for (int k0 = 0; k0 < K; k0 += 4) {
    int ka = k0 + half * 2;
    v2f a, bb;
    a[0] = Wp[(ka + 0) * M];
    a[1] = Wp[(ka + 1) * M];
    bb[0] = nin ? Bp[(ka + 0) * T] : 0.f;
    bb[1] = nin ? Bp[(ka + 1) * T] : 0.f;
    acc = __builtin_amdgcn_wmma_f32_16x16x4_f32(false, a, false, bb,
                                                (short)0, acc, false, false);
  }

  if (!nin) return;  // only after all WMMA issue
#pragma unroll
  for (int r = 0; r < 8; ++r) {
    int m = M0 + r + half * 8;
    int idx = m * T + n;
    float v = acc[r] + bias[m];
    if (MODE == 0) D[idx] = gelu_exact(v);
    else           D[idx] = D[idx] + gamma[m] * v;
  }
}

// ---------------------------------------------------------------------------
// Head conv: cout=1, cin=128, k=7, ctx=6. One thread per output sample.
// ---------------------------------------------------------------------------
__global__ void k_head(const float* __restrict__ xn, const float* __restrict__ cache,
                       const float* __restrict__ w, const float* __restrict__ b,
                       float* __restrict__ out, int T) {
  int t = blockIdx.x * blockDim.x + threadIdx.x;
  if (t >= T) return;
  float acc = b[0];
  for (int c = 0; c < 128; ++c) {
    const float* wr = w + c * 7;
#pragma unroll
    for (int kk = 0; kk < 7; ++kk) {
      int ti = t + kk;
      float v = (ti < 6) ? cache[c * 6 + ti] : xn[c * T + (ti - 6)];
      acc += v * wr[kk];
    }
  }
  out[t] = acc;
}

// ---------------------------------------------------------------------------
// Host orchestration
// ---------------------------------------------------------------------------
static inline void launch_gemm(int mode, const float* W, const float* B, float* D,
                               const float* bias, const float* gamma,
                               int M, int K, int T, hipStream_t stream) {
  if (T % 64 == 0) {
    dim3 g(M / 128, T / 64);
    if (mode == 0) k_gemm_wmma_fast<0><<<g, 128, 0, stream>>>(W, B, D, bias, gamma, M, K, T);
    else           k_gemm_wmma_fast<1><<<g, 128, 0, stream>>>(W, B, D, bias, gamma, M, K, T);
  } else {
    dim3 g(M / 64, (T + 15) / 16);
    if (mode == 0) k_gemm_wmma_ref<0><<<g, 128, 0, stream>>>(W, B, D, bias, gamma, M, K, T);
    else           k_gemm_wmma_ref<1><<<g, 128, 0, stream>>>(W, B, D, bias, gamma, M, K, T);
  }
}

extern "C" void kernel_launch(void* const* d_in, const int* in_sizes, int n_in,
                              void* d_out, int out_size, void* d_ws, size_t ws_size,
                              hipStream_t stream) {
  (void)in_sizes; (void)n_in; (void)out_size; (void)ws_size;
  auto P = [&](int i) { return (const float*)d_in[i]; };

  // d_in layout (JAX pytree leaf order, sorted dict keys):
  // 0 latent(512) 1 cache_in(30976) 2 final_norm_b 3 final_norm_w 4 head_b 5 head_w
  // then per stage: 2 blocks x [conv_b, conv_w, ffn_b1, ffn_b2, ffn_gamma,
  //   ffn_norm_b, ffn_norm_w, ffn_w1, ffn_w2, gamma, norm_b, norm_w], up_b, up_w
  const float* latent  = P(0);
  const float* cache   = P(1);
  const float* fnorm_b = P(2);
  const float* fnorm_w = P(3);
  const float* head_b  = P(4);
  const float* head_w  = P(5);

  float* out_x     = (float*)d_out;           // 25600 samples
  float* out_cache = (float*)d_out + 25600;   // 30976 cache elements

  const int XCAP = 3276800;                   // max C*T (stage 5: 128*25600)
  float* X0 = (float*)d_ws;
  float* X1 = X0 + XCAP;
  float* Hb = X1 + XCAP;                      // max 4C*T = 13.1M floats

  struct StCfg { int type, cin, cout, k, s, ctx; };
  const StCfg ST[6] = {
      {0,   64, 1024,  7, 1, 6},
      {1, 1024,  512, 16, 8, 1},
      {1,  512,  256, 10, 5, 1},
      {1,  256,  256, 10, 5, 1},
      {1,  256,  128,  8, 4, 1},
      {1,  128,  128,  8, 4, 1},
  };

  const float* cur = latent;   // current activation (C,T), channel-major
  float* bufs[2] = {X0, X1};
  int nb = 0;
  int T = 8;
  int off = 0;                 // running cache offset (same for cache_in / cache_out)
  int pi = 6;                  // param cursor

  for (int s = 0; s < 6; ++s) {
    const StCfg& c = ST[s];

    // up-sampler cache slice: last ctx columns of [cache|x] == x[:, T-ctx:]
    {
      int nthr = c.cin * c.ctx;
      k_copy_cache<<<(nthr + 255) / 256, 256, 0, stream>>>(cur, out_cache + off,
                                                           c.cin, T, c.ctx);
    }
    int coff_up = off;
    off += c.cin * c.ctx;

    float* x = bufs[nb];
    nb ^= 1;
    const float* up_b = P(pi + 24);
    const float* up_w = P(pi + 25);
    if (c.type == 0) {
      int nthr = c.cout * T;
      k_conv_up<<<(nthr + 255) / 256, 256, 0, stream>>>(
          cur, cache + coff_up, up_w, up_b, x, c.cin, c.cout, c.k, c.ctx, T);
    } else {
      int Tout = T * c.s;
      int nthr = c.cout * Tout;
      k_convtr<<<(nthr + 255) / 256, 256, 0, stream>>>(
          cur, cache + coff_up, up_w, up_b, x, c.cin, c.cout, c.k, c.s, T, Tout);
      T = Tout;
    }

    float* scr = bufs[nb];     // scratch buffer (xn), distinct from x
    int C = c.cout;

    for (int b = 0; b < 2; ++b) {
      int p = pi + b * 12;
      // LN1 (norm_w = p+11, norm_b = p+10)
      k_ln<<<T, 128, 0, stream>>>(x, P(p + 11), P(p + 10), scr, C, T);
      // block cache slice from xn
      {
        int nthr = C * 6;
        k_copy_cache<<<(nthr + 255) / 256, 256, 0, stream>>>(scr, out_cache + off,
                                                             C, T, 6);
      }
      int coff = off;
      off += C * 6;
      // depthwise conv + gamma + residual (in-place on x)
      {
        int nthr = C * T;
        k_dwconv<<<(nthr + 255) / 256, 256, 0, stream>>>(
            x, scr, cache + coff, P(p + 1), P(p + 0), P(p + 9), C, T);
      }
      // LN2 (ffn_norm_w = p+6, ffn_norm_b = p+5)
      k_ln<<<T, 128, 0, stream>>>(x, P(p + 6), P(p + 5), scr, C, T);
      // FFN up: H = gelu(W1^T * xn + b1)   [W1 = p+7 (C x 4C), b1 = p+2]
      launch_gemm(0, P(p + 7), scr, Hb, P(p + 2), nullptr, 4 * C, C, T, stream);
      // FFN down: x += ffn_gamma * (W2^T * H + b2)  [W2 = p+8, b2 = p+3, g = p+4]
      launch_gemm(1, P(p + 8), Hb, x, P(p + 3), P(p + 4), C, 4 * C, T, stream);
    }
    cur = x;
    pi += 26;
  }

  // final LN + head conv (cache offset lands at 30208)
  float* scr = bufs[nb];
  k_ln<<<T, 128, 0, stream>>>(cur, fnorm_w, fnorm_b, scr, 128, T);
  {
    int nthr = 128 * 6;
    k_copy_cache<<<(nthr + 255) / 256, 256, 0, stream>>>(scr, out_cache + off,
                                                         128, T, 6);
  }
  k_head<<<(T + 255) / 256, 256, 0, stream>>>(scr, cache + off, head_w, head_b,
                                              out_x, T);
}